// LogLikelihoodLossLayer_15101105013071
// MI455X (gfx1250) — compile-verified
//
#include <hip/hip_runtime.h>
#include <math.h>

typedef float v2f __attribute__((ext_vector_type(2)));
typedef float v4f __attribute__((ext_vector_type(4)));
typedef float v8f __attribute__((ext_vector_type(8)));

#define LDSTR 65   // LDS row stride (floats): conflict-free row AND column access

// One wave32 per (b,t) problem: 64x64 SPD Cholesky + triangular solve + logdet.
// Blocked Cholesky NB=16; Schur trailing updates use V_WMMA_F32_16X16X4_F32.
__global__ __launch_bounds__(32)
void chol_nll_kernel(const float* __restrict__ x,
                     const float* __restrict__ mu,
                     const float* __restrict__ sigma,
                     float* __restrict__ ll_out)
{
    __shared__ float S[64 * LDSTR];
    __shared__ float invd[64];

    const int p    = blockIdx.x;
    const int lane = threadIdx.x;      // 0..31 (wave32)
    const int half = lane >> 4;        // 0: lanes 0-15, 1: lanes 16-31
    const int l16  = lane & 15;

    // ---- stream the 16KB covariance tile into LDS (non-temporal: 210MB stream > L2) ----
    const float* sb = sigma + (size_t)p * 4096;
    #pragma unroll
    for (int it = 0; it < 32; ++it) {
        int idx = it * 32 + lane;            // 1024 float4 chunks
        int row = idx >> 4;
        int col = (idx & 15) << 2;
        v4f v = __builtin_nontemporal_load((const v4f*)(sb + row * 64 + col));
        float* dst = &S[row * LDSTR + col];
        dst[0] = v[0]; dst[1] = v[1]; dst[2] = v[2]; dst[3] = v[3];
    }
    __syncthreads();

    // ---- blocked Cholesky (in-place, lower triangle), NB = 16 ----
    for (int bk = 0; bk < 4; ++bk) {
        const int o = bk * 16;

        // (1) unblocked factor of 16x16 diagonal block (cooperative)
        for (int j = 0; j < 16; ++j) {
            if (lane == 0) {
                float d = sqrtf(S[(o + j) * LDSTR + o + j]);
                S[(o + j) * LDSTR + o + j] = d;
                invd[o + j] = 1.0f / d;
            }
            __syncthreads();
            float inv = invd[o + j];
            float lij = 0.0f;
            if (lane > j && lane < 16) {
                lij = S[(o + lane) * LDSTR + o + j] * inv;
                S[(o + lane) * LDSTR + o + j] = lij;
            }
            __syncthreads();
            if (lane > j && lane < 16) {
                for (int q = j + 1; q <= lane; ++q)
                    S[(o + lane) * LDSTR + o + q] -= lij * S[(o + q) * LDSTR + o + j];
            }
            __syncthreads();
        }

        // (2) panel TRSM: rows r in (o+16..63), solve X * Lkk^T = A (per-row, in registers)
        for (int r = o + 16 + lane; r < 64; r += 32) {
            float a[16];
            #pragma unroll
            for (int j = 0; j < 16; ++j) a[j] = S[r * LDSTR + o + j];
            #pragma unroll
            for (int j = 0; j < 16; ++j) {
                float xj = a[j] * invd[o + j];
                a[j] = xj;
                #pragma unroll
                for (int q = j + 1; q < 16; ++q)
                    a[q] -= xj * S[(o + q) * LDSTR + o + j];
            }
            #pragma unroll
            for (int j = 0; j < 16; ++j) S[r * LDSTR + o + j] = a[j];
        }
        __syncthreads();

        // (3) Schur complement trailing update: C(bi,bj) -= L(bi,bk) * L(bj,bk)^T
        //     16x16x16 per block pair -> 4x V_WMMA_F32_16X16X4_F32 (matrix pipe)
        for (int bi = bk + 1; bi < 4; ++bi) {
            for (int bj = bk + 1; bj <= bi; ++bj) {
                const int ro = bi * 16, co = bj * 16, ko = o;
                v8f c;
                #pragma unroll
                for (int v = 0; v < 8; ++v)             // C: M = v + 8*half, N = l16
                    c[v] = S[(ro + v + 8 * half) * LDSTR + co + l16];
#if __has_builtin(__builtin_amdgcn_wmma_f32_16x16x4_f32)
                #pragma unroll
                for (int kk = 0; kk < 16; kk += 4) {
                    v2f a, b;                            // A/B: K = 2*half + vgpr
                    a[0] = -S[(ro + l16) * LDSTR + ko + kk + 2 * half + 0];
                    a[1] = -S[(ro + l16) * LDSTR + ko + kk + 2 * half + 1];
                    b[0] =  S[(co + l16) * LDSTR + ko + kk + 2 * half + 0];
                    b[1] =  S[(co + l16) * LDSTR + ko + kk + 2 * half + 1];
                    // D = (-A) * B + C  ==  C - A*B
                    c = __builtin_amdgcn_wmma_f32_16x16x4_f32(
                            false, a, false, b, (short)0, c, false, false);
                }
#else
                #pragma unroll
                for (int k = 0; k < 16; ++k)
                    #pragma unroll
                    for (int v = 0; v < 8; ++v)
                        c[v] -= S[(ro + v + 8 * half) * LDSTR + ko + k] *
                                S[(co + l16)          * LDSTR + ko + k];
#endif
                #pragma unroll
                for (int v = 0; v < 8; ++v)
                    S[(ro + v + 8 * half) * LDSTR + co + l16] = c[v];
            }
        }
        __syncthreads();
    }

    // ---- log_det = 2 * sum(log diag(L)): each lane owns 2 diagonal entries ----
    float ld = logf(S[lane * LDSTR + lane]) +
               logf(S[(lane + 32) * LDSTR + (lane + 32)]);

    // ---- forward substitution L y = (x - mu); each lane owns rows lane, lane+32 ----
    float d0 = x[(size_t)p * 64 + lane]      - mu[(size_t)p * 64 + lane];
    float d1 = x[(size_t)p * 64 + lane + 32] - mu[(size_t)p * 64 + lane + 32];
    float y0 = 0.0f, y1 = 0.0f;
    #pragma unroll 1
    for (int j = 0; j < 64; ++j) {
        float dj = (j < 32) ? __shfl(d0, j, 32) : __shfl(d1, j - 32, 32);
        float yj = dj * invd[j];
        if (lane == j)      y0 = yj;
        if (lane + 32 == j) y1 = yj;
        float l0 = S[lane * LDSTR + j];
        float l1 = S[(lane + 32) * LDSTR + j];
        if (lane > j)      d0 -= yj * l0;
        if (lane + 32 > j) d1 -= yj * l1;
    }
    float quad = y0 * y0 + y1 * y1;

    #pragma unroll
    for (int off = 16; off > 0; off >>= 1) {
        quad += __shfl_down(quad, off, 32);
        ld   += __shfl_down(ld,   off, 32);
    }
    if (lane == 0) {
        const float LOG2PI = 1.8378770664093453f;      // log(2*pi)
        ll_out[p] = -0.5f * (quad + 64.0f * LOG2PI + 2.0f * ld);
    }
}

// Deterministic mean: single-block reduction over the 12800 per-problem ll's.
__global__ __launch_bounds__(256)
void reduce_nll_kernel(const float* __restrict__ ll, float* __restrict__ out, int n)
{
    __shared__ float sm[256];
    float s = 0.0f;
    for (int i = threadIdx.x; i < n; i += 256) s += ll[i];
    sm[threadIdx.x] = s;
    __syncthreads();
    for (int st = 128; st > 0; st >>= 1) {
        if (threadIdx.x < st) sm[threadIdx.x] += sm[threadIdx.x + st];
        __syncthreads();
    }
    if (threadIdx.x == 0) out[0] = -sm[0] / (float)n;
}

extern "C" void kernel_launch(void* const* d_in, const int* in_sizes, int n_in,
                              void* d_out, int out_size, void* d_ws, size_t ws_size,
                              hipStream_t stream)
{
    const float* x     = (const float*)d_in[0];   // [B,T,D]
    const float* mu    = (const float*)d_in[1];   // [B,T,D]
    const float* sigma = (const float*)d_in[2];   // [B,T,D,D]
    float* ws  = (float*)d_ws;
    float* out = (float*)d_out;

    const int n = in_sizes[0] / 64;               // B*T = 12800 problems

    chol_nll_kernel<<<n, 32, 0, stream>>>(x, mu, sigma, ws);
    reduce_nll_kernel<<<1, 256, 0, stream>>>(ws, out, n);
}